// MonarchMatrix_73615739453553
// MI455X (gfx1250) — compile-verified
//
#include <hip/hip_runtime.h>

typedef __attribute__((ext_vector_type(2))) float v2f;
typedef __attribute__((ext_vector_type(4))) float v4f;
typedef __attribute__((ext_vector_type(8))) float v8f;
typedef int v4i __attribute__((vector_size(16)));   // matches async-LDS builtin param

#define LOGN 12
#define N 4096
#define HALF 2048          // pairs per row per layer
#define ROWS 16            // rows per workgroup
#define RS 4100            // padded LDS row stride (floats); 4100 % 64 == 4 -> conflict-free A loads
#define NTH 512            // 16 waves of 32
#define NWAVE (NTH / 32)
#define NBLK 64
#define BS 64
#define TWOFF (ROWS * RS)                  // 65600 floats
#define SMEM_FLOATS (TWOFF + HALF * 4)     // 73792 floats
#define SMEM_BYTES (SMEM_FLOATS * 4)       // 295168 bytes (<= 320KB WGP LDS)

#define GLOBAL_AS __attribute__((address_space(1)))
#define LDS_AS    __attribute__((address_space(3)))

#if defined(__AMDGCN__) && __has_builtin(__builtin_amdgcn_global_load_async_to_lds_b128)
#define HAS_ASYNC_LOAD 1
#else
#define HAS_ASYNC_LOAD 0
#endif
#if defined(__AMDGCN__) && __has_builtin(__builtin_amdgcn_global_store_async_from_lds_b128)
#define HAS_ASYNC_STORE 1
#else
#define HAS_ASYNC_STORE 0
#endif

__device__ __forceinline__ void wait_async_zero() {
#if defined(__AMDGCN__)
#if __has_builtin(__builtin_amdgcn_s_wait_asynccnt)
  __builtin_amdgcn_s_wait_asynccnt(0);
#elif HAS_ASYNC_LOAD || HAS_ASYNC_STORE
  asm volatile("s_wait_asynccnt 0x0" ::: "memory");
#endif
#endif
}

// One butterfly stage (12 layers) over 16 rows resident in LDS.
// Twiddles for the active layer are staged into LDS (shared by 16 rows).
__device__ __forceinline__ void butterfly_stage(float* __restrict__ h,
                                                float* __restrict__ twl,
                                                const float* __restrict__ twg,
                                                int tid) {
  for (int l = 0; l < LOGN; ++l) {
    const float* layer = twg + (size_t)l * (HALF * 4);
    for (int i = tid; i < HALF; i += NTH) {
      *(v4f*)(twl + 4 * i) = *(const v4f*)(layer + 4 * i);
    }
    __syncthreads();
    const int ls = LOGN - 1 - l;       // log2(stride)
    const int stride = 1 << ls;
    for (int idx = tid; idx < ROWS * HALF; idx += NTH) {
      const int r = idx >> 11;          // / HALF
      const int p = idx & (HALF - 1);   // pair index == twiddle index
      const int i = p & (stride - 1);
      const int g = p >> ls;
      const int topi = (g << (ls + 1)) | i;
      float* row = h + r * RS;
      const float t = row[topi];
      const float b = row[topi + stride];
      const v4f w = *(const v4f*)(twl + 4 * p);   // w00 w01 w10 w11
      row[topi]          = fmaf(w.x, t, w.y * b);
      row[topi + stride] = fmaf(w.z, t, w.w * b);
    }
    __syncthreads();
  }
}

__global__ void __launch_bounds__(NTH)
monarch_fused_kernel(const float* __restrict__ x,
                     const float* __restrict__ Ptw,
                     const float* __restrict__ Qtw,
                     const float* __restrict__ blocks,
                     float* __restrict__ out) {
  extern __shared__ float smem[];
  float* h   = smem;          // ROWS x RS (padded rows)
  float* twl = smem + TWOFF;  // one layer of twiddles

  const int tid = threadIdx.x;
  const size_t rowbase = (size_t)blockIdx.x * ROWS * N;

  // ---- Stage in: 16 rows (256KB) HBM -> LDS ----
#if HAS_ASYNC_LOAD
  // Async Global->LDS: bypasses VGPRs, tracked on ASYNCcnt (32 per wave <= 63).
  for (int i = tid; i < ROWS * N / 4; i += NTH) {
    const int r = i >> 10;             // / (N/4)
    const int c = (i & 1023) << 2;
    __builtin_amdgcn_global_load_async_to_lds_b128(
        (GLOBAL_AS v4i*)(x + rowbase + (size_t)r * N + c),
        (LDS_AS v4i*)(h + r * RS + c), 0, 0);
  }
  wait_async_zero();
#else
  for (int i = tid; i < ROWS * N / 4; i += NTH) {
    const int r = i >> 10;
    const int c = (i & 1023) << 2;
    *(v4f*)(h + r * RS + c) = *(const v4f*)(x + rowbase + (size_t)r * N + c);
  }
#endif
  __syncthreads();

  // ---- Butterfly P ----
  butterfly_stage(h, twl, Ptw, tid);

  // ---- Block-diagonal GEMM: out(16x64) = in(16x64) * blocks[n]^T, fp32 WMMA ----
  const int lane = tid & 31;
  const int wid  = tid >> 5;
  const int m    = lane & 15;   // M (A) / N (B,D) index
  const int kh   = lane >> 4;   // K-half selector
  for (int bi = 0; bi < NBLK / NWAVE; ++bi) {
    const int blk  = bi * NWAVE + wid;   // each wave owns disjoint blocks/columns
    const int col0 = blk * BS;

    // Preload A fragments: 16x4 f32 layout -> lane(m,kh), v[0]=K0/K2, v[1]=K1/K3
    v2f areg[16];
    #pragma unroll
    for (int ks = 0; ks < 16; ++ks) {
      areg[ks] = *(const v2f*)(h + m * RS + col0 + 4 * ks + 2 * kh);
    }

    v8f acc[4] = {};
    const float* bbase = blocks + (size_t)blk * BS * BS;  // blocks[blk][t][s]
    // FULL unroll: constant areg indices -> no v_movrels/M0; 64 WMMAs inline.
    #pragma unroll
    for (int ks = 0; ks < 16; ++ks) {
      #pragma unroll
      for (int nt = 0; nt < 4; ++nt) {
        // B (4x16, K x N): lane(n=m, kh); B[k][n] = blocks[blk][n][k]
        const v2f bfrag =
            *(const v2f*)(bbase + (size_t)(nt * 16 + m) * BS + 4 * ks + 2 * kh);
        acc[nt] = __builtin_amdgcn_wmma_f32_16x16x4_f32(
            false, areg[ks], false, bfrag, (short)0, acc[nt], false, false);
      }
    }

    // Write back D: VGPR j -> row (j + 8*kh), col nt*16 + m. Wave-exclusive columns.
    #pragma unroll
    for (int nt = 0; nt < 4; ++nt) {
      #pragma unroll
      for (int j = 0; j < 8; ++j) {
        h[(j + 8 * kh) * RS + col0 + nt * 16 + m] = acc[nt][j];
      }
    }
  }
  __syncthreads();

  // ---- Butterfly Q ----
  butterfly_stage(h, twl, Qtw, tid);

  // ---- Stage out: LDS -> HBM ----
#if HAS_ASYNC_STORE
  // Async LDS->Global; S_ENDPGM's implicit wait-idle covers ASYNCcnt drain.
  for (int i = tid; i < ROWS * N / 4; i += NTH) {
    const int r = i >> 10;
    const int c = (i & 1023) << 2;
    __builtin_amdgcn_global_store_async_from_lds_b128(
        (GLOBAL_AS v4i*)(out + rowbase + (size_t)r * N + c),
        (LDS_AS v4i*)(h + r * RS + c), 0, 0);
  }
#else
  for (int i = tid; i < ROWS * N / 4; i += NTH) {
    const int r = i >> 10;
    const int c = (i & 1023) << 2;
    *(v4f*)(out + rowbase + (size_t)r * N + c) = *(const v4f*)(h + r * RS + c);
  }
#endif
}

extern "C" void kernel_launch(void* const* d_in, const int* in_sizes, int n_in,
                              void* d_out, int out_size, void* d_ws, size_t ws_size,
                              hipStream_t stream) {
  (void)n_in; (void)d_ws; (void)ws_size; (void)out_size;
  const float* x      = (const float*)d_in[0];
  const float* Ptw    = (const float*)d_in[1];
  const float* Qtw    = (const float*)d_in[2];
  const float* blocks = (const float*)d_in[3];
  float* out = (float*)d_out;

  const int total_rows = in_sizes[0] / N;      // 16384
  const int grid = total_rows / ROWS;          // 1024 workgroups

  // 288KB dynamic LDS (> default 64KB cap): raise the limit every call
  // (no static guards; the call is idempotent and deterministic).
  (void)hipFuncSetAttribute((const void*)monarch_fused_kernel,
                            hipFuncAttributeMaxDynamicSharedMemorySize, SMEM_BYTES);

  monarch_fused_kernel<<<grid, NTH, SMEM_BYTES, stream>>>(x, Ptw, Qtw, blocks, out);
}